// Aggregator_26439818674919
// MI455X (gfx1250) — compile-verified
//
#include <hip/hip_runtime.h>
#include <stdint.h>

// Mean aggregator: out[n, :] = mean_k W[idx[n,k], :]
//   W:   [vocab=100000, 128] f32 (51.2 MB -> resident in 192 MB L2)
//   idx: [n=100000, 10] int
//   out: [n, 128] f32
// Memory-bound embedding gather. Uses the CDNA5 TDM gather engine
// (tensor_load_to_lds, gather_mode=1, 32-bit row indices) with a two-deep
// software pipeline: issue node n+1's DMA into buffer B, s_wait_tensorcnt 2
// (TENSORcnt is in-order) so node n's rows are ready, reduce from LDS, and
// emit non-temporal output stores so the weight table stays hot in L2.

#define K_NBR 10
#define FEAT 128
#define WAVES_PER_BLOCK 8
#define NODES_PER_WAVE 4                      // grid sizing target
#define ROW_BYTES (FEAT * 4)                  // 512 B per embedding row
#define BUF_BYTES (K_NBR * ROW_BYTES)         // 5120 B per pipeline buffer
#define BUF_FLOATS (K_NBR * FEAT)

typedef float v4f  __attribute__((ext_vector_type(4)));
typedef unsigned int u32x4 __attribute__((ext_vector_type(4)));
typedef int i32x4 __attribute__((ext_vector_type(4)));
typedef int i32x8 __attribute__((ext_vector_type(8)));

#if __has_builtin(__builtin_amdgcn_tensor_load_to_lds)
#define HAVE_TDM 1
#else
#define HAVE_TDM 0
#endif

__device__ __forceinline__ void tdm_gather(u32x4 g0, i32x8 g1, i32x4 g2, i32x4 g3) {
#if HAVE_TDM
#if defined(__clang_major__) && (__clang_major__ >= 23)
  i32x8 gz = {0, 0, 0, 0, 0, 0, 0, 0};
  __builtin_amdgcn_tensor_load_to_lds(g0, g1, g2, g3, gz, 0);   // clang-23: 6-arg
#else
  __builtin_amdgcn_tensor_load_to_lds(g0, g1, g2, g3, 0);       // ROCm 7.2: 5-arg
#endif
#endif
}

__device__ __forceinline__ void wait_tensorcnt_le2() {
#if __has_builtin(__builtin_amdgcn_s_wait_tensorcnt)
  __builtin_amdgcn_s_wait_tensorcnt(2);
#else
  asm volatile("s_wait_tensorcnt 0x2" ::: "memory");
#endif
}
__device__ __forceinline__ void wait_tensorcnt_le0() {
#if __has_builtin(__builtin_amdgcn_s_wait_tensorcnt)
  __builtin_amdgcn_s_wait_tensorcnt(0);
#else
  asm volatile("s_wait_tensorcnt 0x0" ::: "memory");
#endif
}

#if HAVE_TDM
// Issue the two gather-mode TDM ops (8 rows + 2 rows) for one node.
// g0c/g1c carry the loop-invariant descriptor words; only lds_addr,
// tile_dim1 and the index groups vary per node.
__device__ __forceinline__ void issue_node(const int* __restrict__ nbr, int node,
                                           unsigned lds_addr, u32x4 g0c, i32x8 g1c) {
  const int* np = nbr + (size_t)node * K_NBR;
  int idx[K_NBR];
#pragma unroll
  for (int j = 0; j < K_NBR; ++j)
    idx[j] = __builtin_amdgcn_readfirstlane(np[j]);   // wave-uniform -> SMEM/SGPR

  u32x4 g0 = g0c;
  i32x8 g1 = g1c;
  i32x4 g2, g3;

  g0[1] = lds_addr;
  g1[4] = 8;                                  // tile_dim1 = #valid indices
  g2[0] = idx[0]; g2[1] = idx[1]; g2[2] = idx[2]; g2[3] = idx[3];
  g3[0] = idx[4]; g3[1] = idx[5]; g3[2] = idx[6]; g3[3] = idx[7];
  tdm_gather(g0, g1, g2, g3);

  g0[1] = lds_addr + 8u * ROW_BYTES;          // rows 8..9 of the tile
  g1[4] = 2;
  g2[0] = idx[8]; g2[1] = idx[9]; g2[2] = 0; g2[3] = 0;
  g3[0] = 0; g3[1] = 0; g3[2] = 0; g3[3] = 0;
  tdm_gather(g0, g1, g2, g3);
}
#endif

__device__ __forceinline__ void reduce_store(const float* __restrict__ buf, int lane,
                                             float* __restrict__ out, int node) {
  v4f acc = {0.f, 0.f, 0.f, 0.f};
  const v4f* b = (const v4f*)buf;
#pragma unroll
  for (int r = 0; r < K_NBR; ++r)
    acc += b[r * (FEAT / 4) + lane];          // ds_load_b128 x10
  acc *= (1.0f / (float)K_NBR);
  v4f* op = (v4f*)(out + (size_t)node * FEAT);
  __builtin_nontemporal_store(acc, op + lane);
}

__global__ __launch_bounds__(256)
void aggregator_mean_tdm(const float* __restrict__ weight,
                         const int*  __restrict__ nbr,
                         float* __restrict__ out,
                         int n_nodes, int vocab)
{
#if HAVE_TDM
  __shared__ float smem[WAVES_PER_BLOCK * 2 * BUF_FLOATS];   // 80 KB: 2 bufs/wave
#endif
  const int lane = (int)(threadIdx.x & 31u);
  const int wv   = __builtin_amdgcn_readfirstlane((int)(threadIdx.x >> 5));
  int node = __builtin_amdgcn_readfirstlane((int)(blockIdx.x * WAVES_PER_BLOCK)) + wv;
  const int stride = (int)(gridDim.x * WAVES_PER_BLOCK);
  if (node >= n_nodes) return;   // wave-uniform

#if HAVE_TDM
  // ---- Loop-invariant D# words (gather mode, 32-bit indices) ----
  const unsigned long long wp = (unsigned long long)(uintptr_t)weight;
  const unsigned lds0 = (unsigned)(uintptr_t)(&smem[0]) + (unsigned)(wv * 2 * BUF_BYTES);

  u32x4 g0c;
  g0c[0] = 0xC0000001u;                                   // count=1|idx32|gather
  g0c[1] = 0;                                             // lds_addr (per issue)
  g0c[2] = (unsigned)(wp & 0xFFFFFFFFull);                // global_addr lo
  g0c[3] = ((unsigned)(wp >> 32) & 0x01FFFFFFu) | (2u << 30);  // hi | type=2

  i32x8 g1c;
  g1c[0] = (int)(2u << 16);                               // data_size = 4 B
  g1c[1] = (int)(((unsigned)FEAT & 0xFFFFu) << 16);       // tensor_dim0 lo16
  g1c[2] = (int)(((unsigned)vocab & 0xFFFFu) << 16);      // dim0 hi=0 | dim1 lo16
  g1c[3] = (int)((((unsigned)vocab >> 16) & 0xFFFFu)      // dim1 hi16
                 | ((unsigned)FEAT << 16));               // tile_dim0 = 128
  g1c[4] = 0;                                             // tile_dim1 (per issue)
  g1c[5] = FEAT;                                          // dim0_stride = 128 elems
  g1c[6] = 0;
  g1c[7] = 0;

  // ---- Two-deep software pipeline over the grid-stride node loop ----
  int parity = 0;
  issue_node(nbr, node, lds0, g0c, g1c);                  // prologue -> buf 0
  for (;;) {
    const int next = node + stride;
    const bool have_next = (next < n_nodes);
    if (have_next) {
      issue_node(nbr, next, lds0 + (unsigned)((parity ^ 1) * BUF_BYTES), g0c, g1c);
      wait_tensorcnt_le2();   // in-order: the 2 ops for `node` have completed
    } else {
      wait_tensorcnt_le0();
    }
    asm volatile("" ::: "memory");   // LDS buffer[parity] now holds 10 rows

    reduce_store(smem + (size_t)(wv * 2 + parity) * BUF_FLOATS, lane, out, node);

    if (!have_next) break;
    node = next;
    parity ^= 1;
  }
#else
  // Fallback: direct vectorized gather (scalar row base + lane*16B), 10 b128
  // loads in flight per wave, grid-stride loop.
  for (; node < n_nodes; node += stride) {
    const int* np = nbr + (size_t)node * K_NBR;
    v4f acc = {0.f, 0.f, 0.f, 0.f};
#pragma unroll
    for (int j = 0; j < K_NBR; ++j) {
      const int ij = __builtin_amdgcn_readfirstlane(np[j]);
      const v4f* row = (const v4f*)(weight + (size_t)ij * FEAT);
      acc += row[lane];
    }
    acc *= (1.0f / (float)K_NBR);
    v4f* op = (v4f*)(out + (size_t)node * FEAT);
    __builtin_nontemporal_store(acc, op + lane);
  }
#endif
}

extern "C" void kernel_launch(void* const* d_in, const int* in_sizes, int n_in,
                              void* d_out, int out_size, void* d_ws, size_t ws_size,
                              hipStream_t stream) {
  const float* weight = (const float*)d_in[0];
  const int*   nbr    = (const int*)d_in[1];
  float*       out    = (float*)d_out;

  const int vocab   = in_sizes[0] / FEAT;      // 100000
  const int n_nodes = out_size / FEAT;         // 100000

  const int nodes_per_block = WAVES_PER_BLOCK * NODES_PER_WAVE;
  dim3 block(32 * WAVES_PER_BLOCK);            // 8 wave32 waves
  dim3 grid((n_nodes + nodes_per_block - 1) / nodes_per_block);   // ~4 nodes/wave
  aggregator_mean_tdm<<<grid, block, 0, stream>>>(weight, nbr, out, n_nodes, vocab);
}